// YoloV1Loss_10316511445008
// MI455X (gfx1250) — compile-verified
//
#include <hip/hip_runtime.h>
#include <hip/hip_bf16.h>

// YOLOv1 loss on MI455X (gfx1250, wave32).
// Memory-bound: ~193 MB streamed once -> NT loads; WMMA f32 16x16x4 used for
// the wave-level reduction (A x ones trick); deterministic 2-pass reduction.

typedef __attribute__((ext_vector_type(2))) float v2f;
typedef __attribute__((ext_vector_type(8))) float v8f;

static constexpr int   BATCH   = 16384;
static constexpr int   CHANS   = 30;
static constexpr int   HW      = 49;            // 7*7
static constexpr int   CELLS   = BATCH * HW;    // 802,816
static constexpr int   THREADS = 256;           // 8 waves (wave32)
static constexpr int   BLOCKS  = 392;           // 392*256*8 == CELLS exactly
static constexpr float INV7    = 1.0f / 7.0f;

// ---------------------------------------------------------------------------
// Wave32 reduction via V_WMMA_F32_16X16X4_F32.
// A-matrix (16x4 f32, 2 VGPRs): lanes 0-15 supply (m=lane, k=0/1),
// lanes 16-31 supply (m=lane-16, k=2/3). Put the lane value in VGPR0, 0 in
// VGPR1, B = all-ones  =>  D[m][n] = v[m] + v[m+16] for every n.
// Summing the 8 D VGPRs gives, on lanes 0-15: sum_{m=0..7}(v[m]+v[m+16]),
// on lanes 16-31: sum_{m=8..15}(v[m]+v[m+16]).  One shfl_down(16) finishes.
// Wave sum is valid on lane 0.
// ---------------------------------------------------------------------------
__device__ __forceinline__ float wave_reduce_wmma(float v) {
  v2f a; a[0] = v;    a[1] = 0.0f;
  v2f b; b[0] = 1.0f; b[1] = 1.0f;
  v8f c = {};
  v8f d = __builtin_amdgcn_wmma_f32_16x16x4_f32(
      /*neg_a=*/false, a, /*neg_b=*/false, b,
      /*c_mod=*/(short)0, c, /*reuse_a=*/false, /*reuse_b=*/false);
  float s = ((d[0] + d[1]) + (d[2] + d[3])) + ((d[4] + d[5]) + (d[6] + d[7]));
  s += __shfl_down(s, 16, 32);
  return s;  // lane 0 holds the wave total
}

// Block (256 threads == 8 wave32) reduction; result valid on threadIdx.x==0.
__device__ __forceinline__ float block_reduce(float v, float* lds8) {
  float ws = wave_reduce_wmma(v);
  const int lane = threadIdx.x & 31;
  const int wid  = threadIdx.x >> 5;
  if (lane == 0) lds8[wid] = ws;
  __syncthreads();
  float r = 0.0f;
  if (threadIdx.x == 0) {
#pragma unroll
    for (int w = 0; w < 8; ++w) r += lds8[w];
  }
  return r;
}

__device__ __forceinline__ float sq(float x) { return x * x; }

// Per-cell YOLOv1 loss. p/l point at (b, 0, r, c); channel k lives at [k*49].
__device__ __forceinline__ float cell_loss(const float* __restrict__ p,
                                           const float* __restrict__ l,
                                           float fc, float fr) {
  const float p0 = __builtin_nontemporal_load(p + 0 * HW);
  const float p1 = __builtin_nontemporal_load(p + 1 * HW);
  const float p2 = __builtin_nontemporal_load(p + 2 * HW);
  const float p3 = __builtin_nontemporal_load(p + 3 * HW);
  const float p4 = __builtin_nontemporal_load(p + 4 * HW);
  const float p5 = __builtin_nontemporal_load(p + 5 * HW);
  const float p6 = __builtin_nontemporal_load(p + 6 * HW);
  const float p7 = __builtin_nontemporal_load(p + 7 * HW);
  const float p8 = __builtin_nontemporal_load(p + 8 * HW);
  const float p9 = __builtin_nontemporal_load(p + 9 * HW);
  const float l0 = __builtin_nontemporal_load(l + 0 * HW);
  const float l1 = __builtin_nontemporal_load(l + 1 * HW);
  const float l2 = __builtin_nontemporal_load(l + 2 * HW);
  const float l3 = __builtin_nontemporal_load(l + 3 * HW);
  const float l4 = __builtin_nontemporal_load(l + 4 * HW);
  const float l5 = __builtin_nontemporal_load(l + 5 * HW);
  const float l6 = __builtin_nontemporal_load(l + 6 * HW);
  const float l7 = __builtin_nontemporal_load(l + 7 * HW);
  const float l8 = __builtin_nontemporal_load(l + 8 * HW);

  // boxes (cx - w/2, cy - h/2, cx + w/2, cy + h/2)
  const float acx = (p0 + fc) * INV7, acy = (p1 + fr) * INV7;
  const float ax1 = acx - 0.5f * p2, ay1 = acy - 0.5f * p3;
  const float ax2 = acx + 0.5f * p2, ay2 = acy + 0.5f * p3;

  const float bcx = (p5 + fc) * INV7, bcy = (p6 + fr) * INV7;
  const float bx1 = bcx - 0.5f * p7, by1 = bcy - 0.5f * p8;
  const float bx2 = bcx + 0.5f * p7, by2 = bcy + 0.5f * p8;

  const float lcx = (l0 + fc) * INV7, lcy = (l1 + fr) * INV7;
  const float lx1 = lcx - 0.5f * l2, ly1 = lcy - 0.5f * l3;
  const float lx2 = lcx + 0.5f * l2, ly2 = lcy + 0.5f * l3;

  const float la = (lx2 - lx1) * (ly2 - ly1);

  // IoU box1 vs label
  float iw1 = fmaxf(fminf(ax2, lx2) - fmaxf(ax1, lx1), 0.0f);
  float ih1 = fmaxf(fminf(ay2, ly2) - fmaxf(ay1, ly1), 0.0f);
  float in1 = iw1 * ih1;
  float iou1 = in1 / ((ax2 - ax1) * (ay2 - ay1) + la - in1 + 1e-10f);

  // IoU box2 vs label
  float iw2 = fmaxf(fminf(bx2, lx2) - fmaxf(bx1, lx1), 0.0f);
  float ih2 = fmaxf(fminf(by2, ly2) - fmaxf(by1, ly1), 0.0f);
  float in2 = iw2 * ih2;
  float iou2 = in2 / ((bx2 - bx1) * (by2 - by1) + la - in2 + 1e-10f);

  const bool choose1 = (iou1 >= iou2);

  const float coord1 = 5.0f * (sq(p0 - l0) + sq(p1 - l1)) +
                       sq(__builtin_sqrtf(p2) - __builtin_sqrtf(l2)) +
                       sq(__builtin_sqrtf(p3) - __builtin_sqrtf(l3));
  const float coord2 = 5.0f * (sq(p5 - l5) + sq(p6 - l6)) +
                       sq(__builtin_sqrtf(p7) - __builtin_sqrtf(l7)) +
                       sq(__builtin_sqrtf(p8) - __builtin_sqrtf(l8));
  const float obj1 = sq(p4 - iou1);
  const float obj2 = sq(p9 - iou2);

  float obj_cell = (choose1 ? coord1 : coord2) +
                   (choose1 ? obj1 : obj2) +
                   0.5f * (choose1 ? obj2 : obj1);

  float cls = 0.0f;
#pragma unroll
  for (int k = 10; k < CHANS; ++k) {
    const float dk = __builtin_nontemporal_load(p + k * HW) -
                     __builtin_nontemporal_load(l + k * HW);
    cls = __builtin_fmaf(dk, dk, cls);
  }
  obj_cell += cls;

  const float noobj_cell = 0.5f * sq(p4 + p9);
  return (l4 == 1.0f) ? obj_cell : noobj_cell;
}

__global__ __launch_bounds__(THREADS)
void yolo_loss_partials(const float* __restrict__ pred,
                        const float* __restrict__ label,
                        float* __restrict__ partials) {
  __shared__ float lds8[8];
  const int tid    = blockIdx.x * THREADS + threadIdx.x;
  const int stride = gridDim.x * THREADS;

  float acc = 0.0f;
  for (int i = tid; i < CELLS; i += stride) {
    const int b  = i / HW;
    const int rc = i - b * HW;
    const int r  = rc / 7;
    const int cc = rc - r * 7;
    const size_t base = (size_t)b * (CHANS * HW) + rc;
    const float* p = pred  + base;
    const float* l = label + base;

    // Prefetch next grid-stride cell base (gfx1250 global_prefetch_b8).
    const int inext = i + stride;
    if (inext < CELLS) {
      const size_t nb = (size_t)(inext / HW) * (CHANS * HW) + (inext % HW);
      __builtin_prefetch(pred + nb, 0, 0);
      __builtin_prefetch(label + nb, 0, 0);
    }

    acc += cell_loss(p, l, (float)cc, (float)r);
  }

  const float bsum = block_reduce(acc, lds8);
  if (threadIdx.x == 0) partials[blockIdx.x] = bsum;
}

__global__ __launch_bounds__(THREADS)
void yolo_loss_finalize(const float* __restrict__ partials,
                        int n, float* __restrict__ out) {
  __shared__ float lds8[8];
  float acc = 0.0f;
  for (int i = threadIdx.x; i < n; i += THREADS) acc += partials[i];
  const float total = block_reduce(acc, lds8);
  if (threadIdx.x == 0) out[0] = total * (1.0f / (float)BATCH);
}

extern "C" void kernel_launch(void* const* d_in, const int* in_sizes, int n_in,
                              void* d_out, int out_size, void* d_ws, size_t ws_size,
                              hipStream_t stream) {
  const float* pred  = (const float*)d_in[0];
  const float* label = (const float*)d_in[1];
  float* out      = (float*)d_out;
  float* partials = (float*)d_ws;   // BLOCKS floats (1568 B)

  yolo_loss_partials<<<BLOCKS, THREADS, 0, stream>>>(pred, label, partials);
  yolo_loss_finalize<<<1, THREADS, 0, stream>>>(partials, BLOCKS, out);
}